// Debayer_8907762172169
// MI455X (gfx1250) — compile-verified
//
#include <hip/hip_runtime.h>

// Bilinear GRBG demosaic, [16,1024,1024] f32 -> [16,3,1024,1024] f32.
// Memory-bound: 64MB in + 192MB out => ~11us floor @ 23.3 TB/s.
// Strategy: async global->LDS halo tiles (CDNA5 ASYNCcnt path), parity
// stencils from LDS, non-temporal b128 stores for the streaming output.

#define IMG_H 1024
#define IMG_W 1024
#define TX 256
#define TY 16
#define LDSW (TX + 8)             // 264 floats pitch (covers x-halo, b128 aligned)
#define LDSH (TY + 2)             // 18 rows (y-halo)
#define NSLOT ((LDSW / 4) * LDSH) // 66*18 = 1188 float4 staging slots

#if defined(__gfx1250__) && __has_builtin(__builtin_amdgcn_global_load_async_to_lds_b128)
#define USE_ASYNC_LDS 1
#endif

typedef float float4v __attribute__((ext_vector_type(4)));
typedef int   vi4     __attribute__((vector_size(16)));
#if defined(__HIP_DEVICE_COMPILE__)
typedef __attribute__((address_space(1))) vi4 gvi4; // global (AS1)
typedef __attribute__((address_space(3))) vi4 lvi4; // LDS (AS3)
#endif

__global__ __launch_bounds__(256) void debayer_grbg_kernel(
    const float* __restrict__ img, float* __restrict__ out)
{
    __shared__ float tile[LDSH][LDSW];

    const int t  = threadIdx.x;
    const int x0 = blockIdx.x * TX;
    const int y0 = blockIdx.y * TY;   // always even (TY=16)
    const int bz = blockIdx.z;
    const float* __restrict__ gimg = img + (size_t)bz * IMG_H * IMG_W;

    // ---- stage halo tile into LDS (zero-fill out-of-bounds => 'SAME' pad) ----
    for (int li = t; li < NSLOT; li += 256) {
        const int row = li / (LDSW / 4);
        const int c4  = li % (LDSW / 4);
        const int gy  = y0 - 1 + row;
        const int gx  = x0 - 4 + 4 * c4;
        float* lp = &tile[row][4 * c4];
        if (gy >= 0 && gy < IMG_H && gx >= 0 && gx < IMG_W) {
            const float* gp = gimg + (size_t)gy * IMG_W + gx;
#if defined(USE_ASYNC_LDS) && defined(__HIP_DEVICE_COMPILE__)
            __builtin_amdgcn_global_load_async_to_lds_b128(
                (gvi4*)gp, (lvi4*)lp, 0, 0);
#else
            *(float4v*)lp = *(const float4v*)gp;
#endif
        } else {
            float4v z = {0.f, 0.f, 0.f, 0.f};
            *(float4v*)lp = z;
        }
    }
#if defined(USE_ASYNC_LDS) && defined(__HIP_DEVICE_COMPILE__)
#if __has_builtin(__builtin_amdgcn_s_wait_asynccnt)
    __builtin_amdgcn_s_wait_asynccnt(0);
#else
    asm volatile("s_wait_asynccnt 0" ::: "memory");
#endif
#endif
    __syncthreads();

    // ---- compute: each thread does a 4-wide pixel group on 4 rows ----
    const int txg  = t & 63;          // x group within tile (64 groups * 4 px)
    const int trow = t >> 6;          // 0..3 row phase
    const int lx0  = 4 * txg + 4;     // tile-local column of pixel 0
    const int gxo  = x0 + 4 * txg;    // global x of pixel 0 (even)
    const bool yodd = (trow & 1) != 0; // image-row parity (y0 even, k*4 even)

    float* __restrict__ rpl = out + (size_t)(bz * 3 + 0) * IMG_H * IMG_W;
    float* __restrict__ gpl = out + (size_t)(bz * 3 + 1) * IMG_H * IMG_W;
    float* __restrict__ bpl = out + (size_t)(bz * 3 + 2) * IMG_H * IMG_W;

#pragma unroll
    for (int k = 0; k < 4; ++k) {
        const int r = trow + 4 * k;           // tile row 0..15
        const int y = y0 + r;
        const float* __restrict__ rU = &tile[r][0];     // image row y-1
        const float* __restrict__ rC = &tile[r + 1][0]; // image row y
        const float* __restrict__ rD = &tile[r + 2][0]; // image row y+1

        float u[6], c[6], d[6];
#pragma unroll
        for (int j = 0; j < 6; ++j) {
            u[j] = rU[lx0 - 1 + j];
            c[j] = rC[lx0 - 1 + j];
            d[j] = rD[lx0 - 1 + j];
        }
        // pixel j has center index j+1 in u/c/d
        float4v R, G, B;
        if (!yodd) {
            // even image row: x even -> G site, x odd -> R site
            R.x = 0.5f * (c[0] + c[2]);
            G.x = c[1];
            B.x = 0.5f * (u[1] + d[1]);

            R.y = c[2];
            G.y = 0.25f * (c[1] + c[3] + u[2] + d[2]);
            B.y = 0.25f * (u[1] + u[3] + d[1] + d[3]);

            R.z = 0.5f * (c[2] + c[4]);
            G.z = c[3];
            B.z = 0.5f * (u[3] + d[3]);

            R.w = c[4];
            G.w = 0.25f * (c[3] + c[5] + u[4] + d[4]);
            B.w = 0.25f * (u[3] + u[5] + d[3] + d[5]);
        } else {
            // odd image row: x even -> B site, x odd -> G site
            B.x = c[1];
            G.x = 0.25f * (c[0] + c[2] + u[1] + d[1]);
            R.x = 0.25f * (u[0] + u[2] + d[0] + d[2]);

            G.y = c[2];
            R.y = 0.5f * (u[2] + d[2]);
            B.y = 0.5f * (c[1] + c[3]);

            B.z = c[3];
            G.z = 0.25f * (c[2] + c[4] + u[3] + d[3]);
            R.z = 0.25f * (u[2] + u[4] + d[2] + d[4]);

            G.w = c[4];
            R.w = 0.5f * (u[4] + d[4]);
            B.w = 0.5f * (c[3] + c[5]);
        }
        const size_t o = (size_t)y * IMG_W + gxo;
        __builtin_nontemporal_store(R, (float4v*)(rpl + o));
        __builtin_nontemporal_store(G, (float4v*)(gpl + o));
        __builtin_nontemporal_store(B, (float4v*)(bpl + o));
    }
}

extern "C" void kernel_launch(void* const* d_in, const int* in_sizes, int n_in,
                              void* d_out, int out_size, void* d_ws, size_t ws_size,
                              hipStream_t stream) {
    (void)n_in; (void)out_size; (void)d_ws; (void)ws_size;
    const float* img = (const float*)d_in[0];
    float* out = (float*)d_out;
    const int bs = in_sizes[0] / (IMG_H * IMG_W);   // 16
    dim3 grid(IMG_W / TX, IMG_H / TY, bs);          // 4 x 64 x 16 = 4096 blocks
    debayer_grbg_kernel<<<grid, dim3(256, 1, 1), 0, stream>>>(img, out);
}